// EquiAttentionBlock_14516989461103
// MI455X (gfx1250) — compile-verified
//
#include <hip/hip_runtime.h>
#include <math.h>

#define NPT 1024
#define HD  128

typedef __attribute__((ext_vector_type(2))) float v2f;
typedef __attribute__((ext_vector_type(8))) float v8f;

// ---------------------------------------------------------------------------
// Multi-tile WMMA helper: accumulate NT adjacent 16x16 output tiles sharing
// one A fragment per K-step (V_WMMA_F32_16X16X4_F32, exact fp32).
// A tile starts at Aptr (row-major, leading dim lda).
// B: if BT, element (kk,n) of tile t is at Bptr[(16*t+n)*ldb + kk]
//    (contiguous (kk,kk+1) pairs -> global_load_b64);
//    else at Bptr[kk*ldb + 16*t + n].
// Fragment layout (ISA 7.12.2, 32-bit 16x4): lane L -> row/col r=L&15,
// K pair base kb = (L<16 ? 0 : 2); VGPR0=K(kb), VGPR1=K(kb+1).
// ---------------------------------------------------------------------------
template <bool BT, int NT>
__device__ inline void wmma_f32_multi(const float* __restrict__ A, int lda,
                                      const float* __restrict__ B, int ldb,
                                      int K, v8f* acc) {
  const int lane  = threadIdx.x & 31;
  const int r     = lane & 15;
  const int kb    = (lane >> 4) << 1;  // 0 or 2
  const int bstep = BT ? 16 * ldb : 16;
  for (int k0 = 0; k0 < K; k0 += 4) {
    v2f a;
    a.x = A[r * lda + k0 + kb];
    a.y = A[r * lda + k0 + kb + 1];
#pragma unroll
    for (int t = 0; t < NT; ++t) {
      const float* Bt = B + t * bstep;
      v2f b;
      if (BT) {
        b.x = Bt[r * ldb + k0 + kb];
        b.y = Bt[r * ldb + k0 + kb + 1];
      } else {
        b.x = Bt[(k0 + kb) * ldb + r];
        b.y = Bt[(k0 + kb + 1) * ldb + r];
      }
      acc[t] = __builtin_amdgcn_wmma_f32_16x16x4_f32(false, a, false, b,
                                                     (short)0, acc[t], false,
                                                     false);
    }
  }
}

// ---------------------------------------------------------------------------
// K1: p[n][h] = x[n,0]*We1[0,h] + x[n,1]*We1[1,h]
// ---------------------------------------------------------------------------
__global__ void k_pre_p(const float* __restrict__ x, const float* __restrict__ We1,
                        float* __restrict__ p) {
  const int n = blockIdx.x;
  const int h = threadIdx.x;  // 128 threads
  p[n * HD + h] = x[n * 2 + 0] * We1[0 * HD + h] + x[n * 2 + 1] * We1[1 * HD + h];
}

// ---------------------------------------------------------------------------
// K2: w2c[h] = sum_o We2[h,o]*Wc[o];  c0 = sum_o be2[o]*Wc[o] + bc
// (exact collapse of the edge-MLP second linear + gate projection)
// ---------------------------------------------------------------------------
__global__ void k_pre_w2c(const float* __restrict__ We2, const float* __restrict__ Wc,
                          const float* __restrict__ be2, const float* __restrict__ bc,
                          float* __restrict__ w2c, float* __restrict__ c0) {
  const int h = threadIdx.x;  // 128 threads
  float s = 0.f;
  for (int o = 0; o < HD; ++o) s += We2[h * HD + o] * Wc[o];
  w2c[h] = s;
  __shared__ float red[HD];
  red[h] = be2[h] * Wc[h];
  __syncthreads();
  for (int st = HD / 2; st > 0; st >>= 1) {
    if (h < st) red[h] += red[h + st];
    __syncthreads();
  }
  if (h == 0) c0[0] = red[0] + bc[0];
}

// ---------------------------------------------------------------------------
// K3: q/k/v = h @ W{q,k,v} + b   (grid: 64 M-tiles x 2 N-strips x 3 matrices,
// one wave per block, 16x64 strip per wave).  For z==2 also emits vT
// (128x1024) so the agg GEMM can consume B with contiguous b64 loads.
// ---------------------------------------------------------------------------
__global__ void k_qkv(const float* __restrict__ hmat,
                      const float* __restrict__ Wq, const float* __restrict__ bq,
                      const float* __restrict__ Wk, const float* __restrict__ bk,
                      const float* __restrict__ Wv, const float* __restrict__ bv,
                      float* __restrict__ q, float* __restrict__ k,
                      float* __restrict__ v, float* __restrict__ vT) {
  const float* W;
  const float* b;
  float* out;
  if (blockIdx.z == 0)      { W = Wq; b = bq; out = q; }
  else if (blockIdx.z == 1) { W = Wk; b = bk; out = k; }
  else                      { W = Wv; b = bv; out = v; }
  const bool isV = (blockIdx.z == 2);

  const int tm = blockIdx.x * 16;
  const int tn = blockIdx.y * 64;
  v8f acc[4] = {};
  wmma_f32_multi<false, 4>(hmat + tm * HD, HD, W + tn, HD, HD, acc);

  const int lane  = threadIdx.x & 31;
  const int col   = lane & 15;
  const int rbase = (lane >> 4) * 8;
#pragma unroll
  for (int t = 0; t < 4; ++t) {
    const int c = tn + 16 * t + col;
    const float bias = b[c];
#pragma unroll
    for (int r2 = 0; r2 < 8; ++r2) {
      const int row = tm + rbase + r2;
      const float val = acc[t][r2] + bias;
      out[row * HD + c] = val;
      if (isV) vT[c * NPT + row] = val;
    }
  }
}

// ---------------------------------------------------------------------------
// K4: S = (q @ k^T) / sqrt(H)   (grid: 64 x 16, 16x64 strip per wave)
// ---------------------------------------------------------------------------
__global__ void k_scores(const float* __restrict__ q, const float* __restrict__ k,
                         float* __restrict__ S) {
  const int tm = blockIdx.x * 16;
  const int tn = blockIdx.y * 64;
  v8f acc[4] = {};
  wmma_f32_multi<true, 4>(q + tm * HD, HD, k + tn * HD, HD, HD, acc);

  const float scale = 0.088388347648318447f;  // 1/sqrt(128)
  const int lane  = threadIdx.x & 31;
  const int col   = lane & 15;
  const int rbase = (lane >> 4) * 8;
#pragma unroll
  for (int t = 0; t < 4; ++t)
#pragma unroll
    for (int r2 = 0; r2 < 8; ++r2)
      S[(tm + rbase + r2) * NPT + tn + 16 * t + col] = acc[t][r2] * scale;
}

// ---------------------------------------------------------------------------
// K5: in-place row softmax over 1024 columns (one block of 256 per row)
// ---------------------------------------------------------------------------
__global__ void k_softmax(float* __restrict__ S) {
  const int row = blockIdx.x;
  float* Sr = S + (size_t)row * NPT;
  const int t = threadIdx.x;
  __shared__ float red[256];

  float m = -INFINITY;
  for (int j = t; j < NPT; j += 256) m = fmaxf(m, Sr[j]);
  red[t] = m;
  __syncthreads();
  for (int st = 128; st > 0; st >>= 1) {
    if (t < st) red[t] = fmaxf(red[t], red[t + st]);
    __syncthreads();
  }
  m = red[0];
  __syncthreads();

  float s = 0.f;
  for (int j = t; j < NPT; j += 256) {
    float e = __expf(Sr[j] - m);
    Sr[j] = e;
    s += e;
  }
  red[t] = s;
  __syncthreads();
  for (int st = 128; st > 0; st >>= 1) {
    if (t < st) red[t] += red[t + st];
    __syncthreads();
  }
  const float inv = 1.0f / red[0];
  for (int j = t; j < NPT; j += 256) Sr[j] *= inv;
}

// ---------------------------------------------------------------------------
// K6: out_h = h + attn @ v    (grid: 64 x 4, 16x32 strip per wave, K = 1024,
// B consumed as vT so both fragments are contiguous b64 loads)
// ---------------------------------------------------------------------------
__global__ void k_agg(const float* __restrict__ attn, const float* __restrict__ vT,
                      const float* __restrict__ hmat, float* __restrict__ out_h) {
  const int tm = blockIdx.x * 16;
  const int tn = blockIdx.y * 32;
  v8f acc[2] = {};
  wmma_f32_multi<true, 2>(attn + (size_t)tm * NPT, NPT, vT + (size_t)tn * NPT,
                          NPT, NPT, acc);

  const int lane  = threadIdx.x & 31;
  const int col   = lane & 15;
  const int rbase = (lane >> 4) * 8;
#pragma unroll
  for (int t = 0; t < 2; ++t)
#pragma unroll
    for (int r2 = 0; r2 < 8; ++r2) {
      const int idx = (tm + rbase + r2) * HD + tn + 16 * t + col;
      out_h[idx] = hmat[idx] + acc[t][r2];
    }
}

// ---------------------------------------------------------------------------
// K7: delta_x[i] = sum_j attn[i,j] * gate(i,j) * (x_i - x_j)
//     gate(i,j) = sum_h relu(p[i][h]+be1[h]-p[j][h]) * w2c[h] + c0
// ---------------------------------------------------------------------------
__global__ void k_delta(const float* __restrict__ x, const float* __restrict__ attn,
                        const float* __restrict__ p, const float* __restrict__ be1,
                        const float* __restrict__ w2c, const float* __restrict__ c0,
                        float* __restrict__ out_x) {
  const int i = blockIdx.x;
  const int t = threadIdx.x;  // 256 threads
  __shared__ float pb[HD];
  __shared__ float wl[HD];
  __shared__ float redx[256];
  __shared__ float redy[256];

  if (t < HD) {
    pb[t] = p[i * HD + t] + be1[t];
    wl[t] = w2c[t];
  }
  __syncthreads();

  const float xi0 = x[i * 2 + 0];
  const float xi1 = x[i * 2 + 1];
  const float C0  = c0[0];

  float dx = 0.f, dy = 0.f;
  for (int j = t; j < NPT; j += 256) {
    // speculative prefetch of the next p-row this lane will touch
    __builtin_prefetch(p + (j + 256) * HD, 0, 0);
    const float w = attn[(size_t)i * NPT + j];
    const float4* pj = (const float4*)(p + j * HD);
    float g = C0;
#pragma unroll 4
    for (int hh = 0; hh < HD / 4; ++hh) {
      const float4 pv = pj[hh];
      g += fmaxf(pb[4 * hh + 0] - pv.x, 0.f) * wl[4 * hh + 0];
      g += fmaxf(pb[4 * hh + 1] - pv.y, 0.f) * wl[4 * hh + 1];
      g += fmaxf(pb[4 * hh + 2] - pv.z, 0.f) * wl[4 * hh + 2];
      g += fmaxf(pb[4 * hh + 3] - pv.w, 0.f) * wl[4 * hh + 3];
    }
    const float wg = w * g;
    dx += wg * (xi0 - x[j * 2 + 0]);
    dy += wg * (xi1 - x[j * 2 + 1]);
  }
  redx[t] = dx;
  redy[t] = dy;
  __syncthreads();
  for (int st = 128; st > 0; st >>= 1) {
    if (t < st) { redx[t] += redx[t + st]; redy[t] += redy[t + st]; }
    __syncthreads();
  }
  if (t == 0) {
    out_x[i * 2 + 0] = xi0 + redx[0];
    out_x[i * 2 + 1] = xi1 + redy[0];
  }
}

// ---------------------------------------------------------------------------
// launch
// ---------------------------------------------------------------------------
extern "C" void kernel_launch(void* const* d_in, const int* in_sizes, int n_in,
                              void* d_out, int out_size, void* d_ws, size_t ws_size,
                              hipStream_t stream) {
  (void)in_sizes; (void)n_in; (void)out_size; (void)ws_size;
  const float* h   = (const float*)d_in[0];
  const float* x   = (const float*)d_in[1];
  // d_in[2] = batch (int64) : unused by the reference math
  const float* Wq  = (const float*)d_in[3];
  const float* bq  = (const float*)d_in[4];
  const float* Wk  = (const float*)d_in[5];
  const float* bk  = (const float*)d_in[6];
  const float* Wv  = (const float*)d_in[7];
  const float* bv  = (const float*)d_in[8];
  const float* We1 = (const float*)d_in[9];
  const float* be1 = (const float*)d_in[10];
  const float* We2 = (const float*)d_in[11];
  const float* be2 = (const float*)d_in[12];
  const float* Wc  = (const float*)d_in[13];
  const float* bc  = (const float*)d_in[14];

  float* out_h = (float*)d_out;             // [N, H]
  float* out_x = out_h + NPT * HD;          // [N, 2]

  // workspace layout (floats): q,k,v,p,vT (N*H each) + w2c + c0 + S (N*N)
  float* ws  = (float*)d_ws;
  float* q   = ws;
  float* k   = q + NPT * HD;
  float* v   = k + NPT * HD;
  float* p   = v + NPT * HD;
  float* vT  = p + NPT * HD;
  float* w2c = vT + NPT * HD;
  float* c0  = w2c + HD;
  float* S   = c0 + 4;

  k_pre_p<<<NPT, HD, 0, stream>>>(x, We1, p);
  k_pre_w2c<<<1, HD, 0, stream>>>(We2, Wc, be2, bc, w2c, c0);
  k_qkv<<<dim3(NPT / 16, 2, 3), 32, 0, stream>>>(h, Wq, bq, Wk, bk, Wv, bv,
                                                 q, k, v, vT);
  k_scores<<<dim3(NPT / 16, NPT / 64), 32, 0, stream>>>(q, k, S);
  k_softmax<<<NPT, 256, 0, stream>>>(S);
  k_agg<<<dim3(NPT / 16, HD / 32), 32, 0, stream>>>(S, vT, h, out_h);
  k_delta<<<NPT, 256, 0, stream>>>(x, S, p, be1, w2c, c0, out_x);
}